// GNNModule_46531675685433
// MI455X (gfx1250) — compile-verified
//
#include <hip/hip_runtime.h>

typedef __attribute__((ext_vector_type(2))) float v2f;
typedef __attribute__((ext_vector_type(8))) float v8f;
typedef __attribute__((ext_vector_type(4))) unsigned int v4u;
typedef __attribute__((ext_vector_type(4))) int v4i;
typedef __attribute__((ext_vector_type(8))) int v8i;

#define N_NODES   10000
#define E_EDGES   160000
#define E_TOT     (E_EDGES + N_NODES)
#define G_GRAPHS  64
#define C_CLASSES 6
#define NEG_SLOPE 0.2f
#define EPS_F     1e-16f

// ---------------------------------------------------------------------------
// GEMM: H[M x Nout] = X[M x K] @ W[K x Nout]   (row-major fp32)
// 256 threads = 8 wave32 waves; block tile 64x64, K-step 16.
// A/B tiles staged to LDS by the Tensor Data Mover (double-buffered: DMA of
// tile k+1 overlaps WMMA of tile k). TDM zero-fills OOB (ragged M / Nout=6)
// and inserts LDS padding for conflict-free fragment reads.
//   A tile: 64 rows x 16 k, LDS stride 20 dwords (pad 16+4)
//   B tile: 16 k   x 64 c,  LDS stride 72 dwords (pad 64+8)
// ---------------------------------------------------------------------------
#define BM   64
#define BN   64
#define BKK  16
#define LDA  20   // A LDS row stride (dwords)
#define LDB  72   // B LDS row stride (dwords)

#define HAVE_TDM (__has_builtin(__builtin_amdgcn_tensor_load_to_lds) && \
                  __has_builtin(__builtin_amdgcn_s_wait_tensorcnt))

#if HAVE_TDM
// Issue one 2D TDM tile load: global (elems, 4B each) -> LDS with padding.
__device__ __forceinline__ void tdm_load_2d(unsigned lds_addr, const void* gptr,
                                            unsigned tensor_d0, unsigned tensor_d1,
                                            unsigned d0_stride,
                                            unsigned tile_d0, unsigned tile_d1,
                                            unsigned pad_int_code, unsigned pad_amt_code) {
  unsigned long long ga = (unsigned long long)(uintptr_t)gptr;
  v4u g0;
  g0[0] = 1u;                                     // count=1 (valid user D#)
  g0[1] = lds_addr;                               // lds_addr [63:32]
  g0[2] = (unsigned)(ga & 0xFFFFFFFFu);           // global_addr[31:0]
  g0[3] = (unsigned)((ga >> 32) & 0x01FFFFFFu) |  // global_addr[56:32]
          (2u << 30);                              // type=2 ("image")
  v8i g1;
  g1[0] = (int)((2u << 16)                        // data_size = 4B
              | (1u << 20)                        // pad_enable
              | (pad_int_code << 22)              // pad_interval
              | (pad_amt_code << 25));            // pad_amount
  g1[1] = (int)((tensor_d0 & 0xFFFFu) << 16);     // tensor_dim0[15:0] @63:48
  g1[2] = (int)(((tensor_d0 >> 16) & 0xFFFFu) | ((tensor_d1 & 0xFFFFu) << 16));
  g1[3] = (int)(((tensor_d1 >> 16) & 0xFFFFu) | ((tile_d0 & 0xFFFFu) << 16));
  g1[4] = (int)(tile_d1 & 0xFFFFu);               // tile_dim1; tile_dim2=0
  g1[5] = (int)d0_stride;                         // tensor_dim0_stride[31:0]
  g1[6] = 0;                                      // stride[47:32], dim1_stride lo
  g1[7] = 0;
  v4i gz = {0, 0, 0, 0};
#if __clang_major__ >= 23
  v8i gz8 = {0, 0, 0, 0, 0, 0, 0, 0};
  __builtin_amdgcn_tensor_load_to_lds(g0, g1, gz, gz, gz8, 0);
#else
  __builtin_amdgcn_tensor_load_to_lds(g0, g1, gz, gz, 0);
#endif
}
#endif

__global__ __launch_bounds__(256)
void gemm_xw_wmma(const float* __restrict__ X, const float* __restrict__ W,
                  float* __restrict__ H, int M, int K, int Nout) {
  __shared__ __align__(16) float As[2][BM * LDA + 16];
  __shared__ __align__(16) float Bs[2][BKK * LDB + 16];

  const int tid  = threadIdx.x;
  const int lane = tid & 31;
  const int wave = tid >> 5;
  const int half = lane >> 4;
  const int lcol = lane & 15;

  const int row0 = blockIdx.y * BM;
  const int col0 = blockIdx.x * BN;
  const int wr = wave & 3;                  // 16-row strip
  const int wc = wave >> 2;                 // 32-col strip
  const int wrow = row0 + wr * 16;
  const int wcol = col0 + wc * 32;

  v8f acc0 = {0.f,0.f,0.f,0.f,0.f,0.f,0.f,0.f};
  v8f acc1 = acc0;

#if HAVE_TDM
  if (tid < 32) {   // one wave issues the DMA (TDM ignores EXEC)
    tdm_load_2d((unsigned)(uintptr_t)&As[0][0], X + (size_t)row0 * K,
                (unsigned)K, (unsigned)(M - row0), (unsigned)K,
                BKK, BM, /*pad 16 dw*/3, /*amt 4 dw*/3);
    tdm_load_2d((unsigned)(uintptr_t)&Bs[0][0], W + col0,
                (unsigned)(Nout - col0), (unsigned)K, (unsigned)Nout,
                BN, BKK, /*pad 64 dw*/5, /*amt 8 dw*/7);
  }
#else
  {
    int r  = tid >> 2, kk = (tid & 3) * 4;
    float4 av = make_float4(0.f, 0.f, 0.f, 0.f);
    if (row0 + r < M) av = *(const float4*)(X + (size_t)(row0 + r) * K + kk);
    *(float4*)&As[0][r * LDA + kk] = av;
    int bk = tid >> 4, c4 = (tid & 15) * 4;
    for (int j = 0; j < 4; ++j) {
      int gc = col0 + c4 + j;
      Bs[0][bk * LDB + c4 + j] = (gc < Nout) ? W[(size_t)bk * Nout + gc] : 0.f;
    }
  }
#endif

  int buf = 0;
  for (int kb = 0; kb < K; kb += BKK) {
    const bool more = (kb + BKK) < K;
#if HAVE_TDM
    if (tid < 32) {
      if (more) {   // prefetch next tile pair into the other buffer
        tdm_load_2d((unsigned)(uintptr_t)&As[buf ^ 1][0],
                    X + (size_t)row0 * K + (kb + BKK),
                    (unsigned)(K - kb - BKK), (unsigned)(M - row0), (unsigned)K,
                    BKK, BM, 3, 3);
        tdm_load_2d((unsigned)(uintptr_t)&Bs[buf ^ 1][0],
                    W + (size_t)(kb + BKK) * Nout + col0,
                    (unsigned)(Nout - col0), (unsigned)(K - kb - BKK), (unsigned)Nout,
                    BN, BKK, 5, 7);
        __builtin_amdgcn_s_wait_tensorcnt(2);   // current pair landed
      } else {
        __builtin_amdgcn_s_wait_tensorcnt(0);
      }
    }
#else
    if (more) {
      int r  = tid >> 2, kk = (tid & 3) * 4;
      float4 av = make_float4(0.f, 0.f, 0.f, 0.f);
      if (row0 + r < M)
        av = *(const float4*)(X + (size_t)(row0 + r) * K + (kb + BKK) + kk);
      *(float4*)&As[buf ^ 1][r * LDA + kk] = av;
      int bk = tid >> 4, c4 = (tid & 15) * 4;
      for (int j = 0; j < 4; ++j) {
        int gc = col0 + c4 + j;
        Bs[buf ^ 1][bk * LDB + c4 + j] =
            (gc < Nout) ? W[(size_t)(kb + BKK + bk) * Nout + gc] : 0.f;
      }
    }
#endif
    __syncthreads();

    const float* Ab = &As[buf][0];
    const float* Bb = &Bs[buf][0];
#pragma unroll
    for (int kk = 0; kk < BKK; kk += 4) {
      // A 16x4 frag: lane holds A[M=lcol][K = kk + 2*half + j]
      v2f a = *(const v2f*)&Ab[(wr * 16 + lcol) * LDA + kk + 2 * half];
      // B 4x16 frag: lane holds B[K = kk + 2*half + j][N = col]
      int k0 = kk + 2 * half;
      v2f b0, b1;
      b0[0] = Bb[(k0    ) * LDB + wc * 32 + lcol];
      b0[1] = Bb[(k0 + 1) * LDB + wc * 32 + lcol];
      b1[0] = Bb[(k0    ) * LDB + wc * 32 + 16 + lcol];
      b1[1] = Bb[(k0 + 1) * LDB + wc * 32 + 16 + lcol];
      acc0 = __builtin_amdgcn_wmma_f32_16x16x4_f32(false, a, false, b0,
                                                   (short)0, acc0, false, false);
      acc1 = __builtin_amdgcn_wmma_f32_16x16x4_f32(false, a, false, b1,
                                                   (short)0, acc1, false, false);
    }
    __syncthreads();
    buf ^= 1;
  }

  // C/D layout: VGPR v -> row (v + 8*half), col lcol
#pragma unroll
  for (int v = 0; v < 8; ++v) {
    int grow = wrow + v + 8 * half;
    if (grow >= M) continue;
    int gc0 = wcol + lcol;
    int gc1 = wcol + 16 + lcol;
    if (gc0 < Nout) H[(size_t)grow * Nout + gc0] = acc0[v];
    if (gc1 < Nout) H[(size_t)grow * Nout + gc1] = acc1[v];
  }
}

// ---------------------------------------------------------------------------
// Per-node attention scores: s_src[n] = h[n].a_src, s_dst[n] = h[n].a_dst
// ---------------------------------------------------------------------------
__global__ void attn_scores(const float* __restrict__ H,
                            const float* __restrict__ a_src,
                            const float* __restrict__ a_dst,
                            float* __restrict__ s_src, float* __restrict__ s_dst,
                            int n, int F) {
  int node = blockIdx.x * (blockDim.x >> 5) + (threadIdx.x >> 5);
  int lane = threadIdx.x & 31;
  if (node >= n) return;
  const float* h = H + (size_t)node * F;
  float ds = 0.f, dd = 0.f;
  for (int f = lane; f < F; f += 32) {
    float v = h[f];
    ds += v * a_src[f];
    dd += v * a_dst[f];
  }
#pragma unroll
  for (int off = 16; off > 0; off >>= 1) {
    ds += __shfl_down(ds, off, 32);
    dd += __shfl_down(dd, off, 32);
  }
  if (lane == 0) { s_src[node] = ds; s_dst[node] = dd; }
}

__global__ void init_softmax(float* __restrict__ m, float* __restrict__ denom, int n) {
  int i = blockIdx.x * blockDim.x + threadIdx.x;
  if (i < n) { ((unsigned int*)m)[i] = 0xFF800000u; denom[i] = 0.f; }  // -inf, 0
}

__device__ __forceinline__ int edge_src(const int* ei, int e) {
  return (e < E_EDGES) ? ei[e] : (e - E_EDGES);
}
__device__ __forceinline__ int edge_dst(const int* ei, int e) {
  return (e < E_EDGES) ? ei[E_EDGES + e] : (e - E_EDGES);
}

// Pass 1: leaky-relu logits + segment max (float atomic max via int/uint trick)
__global__ void edge_logits(const int* __restrict__ ei,
                            const float* __restrict__ s_src,
                            const float* __restrict__ s_dst,
                            float* __restrict__ alpha, float* __restrict__ m) {
  int e = blockIdx.x * blockDim.x + threadIdx.x;
  if (e >= E_TOT) return;
  int s = edge_src(ei, e), d = edge_dst(ei, e);
  float a = s_src[s] + s_dst[d];
  a = (a > 0.f) ? a : NEG_SLOPE * a;
  alpha[e] = a;
  if (a >= 0.f) atomicMax((int*)(m + d), __float_as_int(a));
  else          atomicMin((unsigned int*)(m + d), __float_as_uint(a));
}

// Pass 2: exponentiate (in place) + segment sum
__global__ void edge_expsum(const int* __restrict__ ei, float* __restrict__ alpha,
                            const float* __restrict__ m, float* __restrict__ denom) {
  int e = blockIdx.x * blockDim.x + threadIdx.x;
  if (e >= E_TOT) return;
  int d = edge_dst(ei, e);
  float ex = __expf(alpha[e] - m[d]);
  alpha[e] = ex;
  atomicAdd(denom + d, ex);
}

// Pass 3: normalized scatter-aggregate, one wave per edge (coalesced rows,
// L2-side global_atomic_add_f32).
__global__ void edge_scatter(const int* __restrict__ ei, const float* __restrict__ ex,
                             const float* __restrict__ denom,
                             const float* __restrict__ H, float* __restrict__ Y, int F) {
  int e    = blockIdx.x * (blockDim.x >> 5) + (threadIdx.x >> 5);
  int lane = threadIdx.x & 31;
  if (e >= E_TOT) return;
  int s = edge_src(ei, e), d = edge_dst(ei, e);
  float coef = ex[e] / (denom[d] + EPS_F);
  const float* hs = H + (size_t)s * F;
  float* yd = Y + (size_t)d * F;
  for (int f = lane; f < F; f += 32) atomicAdd(yd + f, coef * hs[f]);
}

__global__ void bias_relu(float* __restrict__ Y, const float* __restrict__ bias,
                          int total, int F) {
  int i = blockIdx.x * blockDim.x + threadIdx.x;
  if (i >= total) return;
  float v = Y[i] + bias[i % F];
  Y[i] = (v > 0.f) ? v : 0.f;
}

__global__ void pool_graphs(const float* __restrict__ Y, const int* __restrict__ batch,
                            float* __restrict__ gsum, float* __restrict__ gcnt) {
  int n = blockIdx.x * blockDim.x + threadIdx.x;
  if (n >= N_NODES) return;
  int b = batch[n];
#pragma unroll
  for (int c = 0; c < C_CLASSES; ++c)
    atomicAdd(gsum + b * C_CLASSES + c, Y[(size_t)n * C_CLASSES + c]);
  atomicAdd(gcnt + b, 1.0f);
}

__global__ void pool_finalize(const float* __restrict__ gsum,
                              const float* __restrict__ gcnt, float* __restrict__ out) {
  int i = threadIdx.x;
  if (i < G_GRAPHS * C_CLASSES) out[i] = gsum[i] / fmaxf(gcnt[i / C_CLASSES], 1.0f);
}

// ---------------------------------------------------------------------------
extern "C" void kernel_launch(void* const* d_in, const int* in_sizes, int n_in,
                              void* d_out, int out_size, void* d_ws, size_t ws_size,
                              hipStream_t stream) {
  (void)in_sizes; (void)n_in; (void)out_size; (void)ws_size;
  // input order: x, [W,att_src,att_dst,bias] x3, edge_index, batch_index
  const float* x       = (const float*)d_in[0];
  const float* W[3]    = {(const float*)d_in[1], (const float*)d_in[5],  (const float*)d_in[9]};
  const float* asrc[3] = {(const float*)d_in[2], (const float*)d_in[6],  (const float*)d_in[10]};
  const float* adst[3] = {(const float*)d_in[3], (const float*)d_in[7],  (const float*)d_in[11]};
  const float* bias[3] = {(const float*)d_in[4], (const float*)d_in[8],  (const float*)d_in[12]};
  const int*   ei      = (const int*)d_in[13];
  const int*   batch   = (const int*)d_in[14];
  float*       out     = (float*)d_out;

  float* ws = (float*)d_ws;
  const size_t NB = (size_t)N_NODES * 512;
  float* bufA  = ws;            // h (pre-aggregation), reused each layer
  float* bufB  = ws + NB;       // y0, then y2
  float* bufC  = ws + 2 * NB;   // y1
  float* s_src = ws + 3 * NB;
  float* s_dst = s_src + N_NODES;
  float* m     = s_dst + N_NODES;
  float* denom = m + N_NODES;
  float* alpha = denom + N_NODES;         // E_TOT entries
  float* gsum  = alpha + E_TOT;           // G*C
  float* gcnt  = gsum + G_GRAPHS * C_CLASSES;

  const int    Fin[3]  = {128, 256, 512};
  const int    Fout[3] = {256, 512, 6};
  const float* Xin[3]  = {x, bufB, bufC};
  float*       Yout[3] = {bufB, bufC, bufB};

  for (int i = 0; i < 3; ++i) {
    const int K = Fin[i], F = Fout[i];
    const float* X = Xin[i];
    float* H = bufA;
    float* Y = Yout[i];

    dim3 gg((F + BN - 1) / BN, (N_NODES + BM - 1) / BM);
    gemm_xw_wmma<<<gg, 256, 0, stream>>>(X, W[i], H, N_NODES, K, F);

    attn_scores<<<(N_NODES + 7) / 8, 256, 0, stream>>>(H, asrc[i], adst[i],
                                                       s_src, s_dst, N_NODES, F);
    init_softmax<<<(N_NODES + 255) / 256, 256, 0, stream>>>(m, denom, N_NODES);
    edge_logits<<<(E_TOT + 255) / 256, 256, 0, stream>>>(ei, s_src, s_dst, alpha, m);
    edge_expsum<<<(E_TOT + 255) / 256, 256, 0, stream>>>(ei, alpha, m, denom);

    hipMemsetAsync(Y, 0, (size_t)N_NODES * F * sizeof(float), stream);
    edge_scatter<<<(E_TOT + 7) / 8, 256, 0, stream>>>(ei, alpha, denom, H, Y, F);

    int total = N_NODES * F;
    bias_relu<<<(total + 255) / 256, 256, 0, stream>>>(Y, bias[i], total, F);
  }

  hipMemsetAsync(gsum, 0, (G_GRAPHS * C_CLASSES + G_GRAPHS) * sizeof(float), stream);
  pool_graphs<<<(N_NODES + 255) / 256, 256, 0, stream>>>(Yout[2], batch, gsum, gcnt);
  pool_finalize<<<1, 384, 0, stream>>>(gsum, gcnt, out);
}